// roiPooling_15556371546258
// MI455X (gfx1250) — compile-verified
//
#include <hip/hip_runtime.h>

// RoI pooling (bilinear crop-resize to 7x7) for MI455X / gfx1250.
// Bandwidth-bound: ~144 MB HBM traffic floor -> ~6.2 us at 23.3 TB/s.
// - CDNA5 async gather engine (global_load_async_to_lds_b128 + s_wait_asynccnt)
//   double-buffers the 4 corner channel-vectors per output point through LDS.
// - Output stores are non-temporal b128 (write-once stream) so the 41 MB image
//   and its ~411 MB of L2-side corner-gather reuse keep L2 residency.

#define POOL_P   7
#define IMG_H    100
#define IMG_W    100
#define IMG_C    1024
#define THREADS  256   // 256 threads * 4 channels (float4) = 1024 channels

typedef float v4f __attribute__((ext_vector_type(4)));  // native vector for NT store

__device__ __forceinline__ int iclamp(int v, int lo, int hi) {
    return v < lo ? lo : (v > hi ? hi : v);
}
__device__ __forceinline__ int imin(int a, int b) { return a < b ? a : b; }

struct XInfo { int ax0, ax1; float wx; };

__device__ __forceinline__ XInfo xinfo(int px, int x, int w) {
    // Matches reference: in_x = px * (float(w)/7); floor; x1=min(x0+1,w-1); clamp to image.
    float in_x = (float)px * ((float)w / 7.0f);
    int   x0   = (int)in_x;               // in_x >= 0 -> trunc == floor
    float wx   = in_x - (float)x0;
    int   x1   = imin(x0 + 1, w - 1);
    XInfo r;
    r.ax0 = iclamp(x + x0, 0, IMG_W - 1);
    r.ax1 = iclamp(x + x1, 0, IMG_W - 1);
    r.wx  = wx;
    return r;
}

__device__ __forceinline__ float blerp(float v00, float v01, float v10, float v11,
                                       float wx, float wy) {
    float top = v00 + (v01 - v00) * wx;
    float bot = v10 + (v11 - v10) * wx;
    return top + (bot - top) * wy;
}

// Issue one async 16B-per-lane gather: copies from the global address in gaddr to the LDS offset in ldsoff
#define ASYNC_GATHER_B128(ldsoff, gaddr)                                     \
    asm volatile("global_load_async_to_lds_b128 %0, %1, off"                 \
                 :: "v"(ldsoff), "v"(gaddr) : "memory")

__global__ __launch_bounds__(THREADS)
void roi_pool_kernel(const float* __restrict__ img,
                     const float* __restrict__ rois,
                     float* __restrict__ out)
{
    const int r   = blockIdx.x;      // ROI index
    const int py  = blockIdx.y;      // output row 0..6
    const int tid = threadIdx.x;
    const int c   = tid * 4;         // this thread's channel base

    // ROI (x, y, w, h) — stored as exact small ints in float.
    const int x = (int)rois[r * 4 + 0];
    const int y = (int)rois[r * 4 + 1];
    const int w = (int)rois[r * 4 + 2];
    const int h = (int)rois[r * 4 + 3];

    // Vertical sampling for this output row (uniform across block).
    float in_y = (float)py * ((float)h / 7.0f);
    int   y0   = (int)in_y;
    float wy   = in_y - (float)y0;
    int   y1   = imin(y0 + 1, h - 1);
    int   ay0  = iclamp(y + y0, 0, IMG_H - 1);
    int   ay1  = iclamp(y + y1, 0, IMG_H - 1);

    const float* row0 = img + (size_t)ay0 * (IMG_W * IMG_C);
    const float* row1 = img + (size_t)ay1 * (IMG_W * IMG_C);

    float* outp = out + ((size_t)(r * POOL_P + py) * POOL_P) * IMG_C + c;

    // Double-buffered LDS staging: 2 stages x 4 corners x 256 lanes x 16B = 32 KB.
    __shared__ v4f buf[2][4][THREADS];

    // Per-lane LDS byte offsets (generic LDS pointer low 32 bits == LDS offset).
    unsigned lds[2][4];
#pragma unroll
    for (int s = 0; s < 2; ++s)
#pragma unroll
        for (int k = 0; k < 4; ++k)
            lds[s][k] = (unsigned)(unsigned long long)&buf[s][k][tid];

    // Issue the 4 corner gathers for one pool point into LDS stage `s`.
    auto issue = [&](int s, const XInfo& xi) {
        const float* p00 = row0 + xi.ax0 * IMG_C + c;
        const float* p01 = row0 + xi.ax1 * IMG_C + c;
        const float* p10 = row1 + xi.ax0 * IMG_C + c;
        const float* p11 = row1 + xi.ax1 * IMG_C + c;
        ASYNC_GATHER_B128(lds[s][0], p00);
        ASYNC_GATHER_B128(lds[s][1], p01);
        ASYNC_GATHER_B128(lds[s][2], p10);
        ASYNC_GATHER_B128(lds[s][3], p11);
    };

    XInfo xi = xinfo(0, x, w);
    issue(0, xi);              // prologue: stage 0 in flight (ASYNCcnt = 4)

    int s = 0;
#pragma unroll
    for (int px = 0; px < POOL_P; ++px) {
        const float wx = xi.wx;

        if (px + 1 < POOL_P) {
            XInfo nxt = xinfo(px + 1, x, w);
            issue(s ^ 1, nxt); // prefetch next point (ASYNCcnt = 8)
            xi = nxt;
            // Async loads complete in order: <=4 outstanding means the
            // current stage's 4 gathers have landed in LDS.
            asm volatile("s_wait_asynccnt 0x4" ::: "memory");
        } else {
            asm volatile("s_wait_asynccnt 0x0" ::: "memory");
        }

        // Each lane reads back only its own LDS slots -> no barrier needed.
        // The blerp consumes these (forcing the DScnt wait) before the next
        // iteration's async issue can rewrite this stage.
        v4f v00 = buf[s][0][tid];
        v4f v01 = buf[s][1][tid];
        v4f v10 = buf[s][2][tid];
        v4f v11 = buf[s][3][tid];

        v4f o;
        o.x = blerp(v00.x, v01.x, v10.x, v11.x, wx, wy);
        o.y = blerp(v00.y, v01.y, v10.y, v11.y, wx, wy);
        o.z = blerp(v00.z, v01.z, v10.z, v11.z, wx, wy);
        o.w = blerp(v00.w, v01.w, v10.w, v11.w, wx, wy);

        // Non-temporal 128-bit store: output is write-once, keep L2 for the image.
        __builtin_nontemporal_store(o, (v4f*)(outp + (size_t)px * IMG_C));
        s ^= 1;
    }
}

extern "C" void kernel_launch(void* const* d_in, const int* in_sizes, int n_in,
                              void* d_out, int out_size, void* d_ws, size_t ws_size,
                              hipStream_t stream) {
    const float* img  = (const float*)d_in[0];   // [1,100,100,1024] f32
    const float* rois = (const float*)d_in[1];   // [1,512,4] f32
    float*       out  = (float*)d_out;           // [1,512,7,7,1024] f32

    const int num_rois = in_sizes[1] / 4;        // 512
    dim3 grid(num_rois, POOL_P);
    roi_pool_kernel<<<grid, THREADS, 0, stream>>>(img, rois, out);
}